// VectorQuantizer_36223754175111
// MI455X (gfx1250) — compile-verified
//
#include <hip/hip_runtime.h>
#include <hip/hip_bf16.h>

typedef __attribute__((ext_vector_type(2))) float v2f;
typedef __attribute__((ext_vector_type(4))) float v4f;
typedef __attribute__((ext_vector_type(8))) float v8f;

#define NUM_E      512
#define DIM        64
#define CHUNK      128
#define NT_PER_CH  (CHUNK / 16)
#define CBP_STRIDE 144   // float2 units; 2*144 % 64 == 32 -> halves hit disjoint bank sets
#define WAVES_WG   8
#define TOK_WG     (WAVES_WG * 16)
#define N_TOKENS_C 262144
#define GRID_MAIN  (N_TOKENS_C / TOK_WG)   // 2048

// ---------------------------------------------------------------- prep ----
__global__ __launch_bounds__(512) void vq_prep(const float* __restrict__ cb,
                                               float* __restrict__ e2) {
    const int k = threadIdx.x;              // 0..511
    const float* row = cb + (size_t)k * DIM;
    float s = 0.f;
#pragma unroll
    for (int d = 0; d < DIM; ++d) s = fmaf(row[d], row[d], s);
    e2[k] = s;
}

// ---------------------------------------------------------------- main ----
__global__ __launch_bounds__(256) void vq_main(const float* __restrict__ z,
                                               const float* __restrict__ cb,
                                               const float* __restrict__ e2g,
                                               float* __restrict__ out,
                                               float* __restrict__ partials) {
    // K-pair layout: cbP[p*STRIDE + n] = (cb[n][2p], cb[n][2p+1]), p = K/2 in 0..31
    // -> each B fragment is ONE contiguous ds_load_b64 into an even VGPR pair
    __shared__ v2f   cbP[32 * CBP_STRIDE];  // 36,864 B
    __shared__ float e2s[NUM_E];
    __shared__ float wsum[WAVES_WG];

    const int tid  = threadIdx.x;
    const int lane = tid & 31;
    const int wave = tid >> 5;
    const int half = lane >> 4;             // 0: K%4 in {0,1}   1: K%4 in {2,3}
    const int lcol = lane & 15;
    const int tokBase = (blockIdx.x * WAVES_WG + wave) * 16;

    for (int i = tid; i < NUM_E; i += 256) e2s[i] = e2g[i];

    // ---- A fragments: 16 tokens x 64 dims, ISA 16x4 fp32 layout ----
    v2f a[16];
    const float* zr = z + (size_t)(tokBase + lcol) * DIM + half * 2;
#pragma unroll
    for (int j = 0; j < 16; ++j) a[j] = *(const v2f*)(zr + j * 4);

    float mv[8];
    int   mi[8];
#pragma unroll
    for (int r = 0; r < 8; ++r) { mv[r] = 3.4e38f; mi[r] = 0; }

    for (int cbase = 0; cbase < NUM_E; cbase += CHUNK) {
        __syncthreads();
        // stage chunk as K-pairs; lanes walk ROWS -> b64 store banks
        // (32p + 2*row) % 64 cover all 64 banks: conflict-free
        for (int i = 0; i < (CHUNK * DIM / 4) / 256; ++i) {
            const int g   = i * 256 + tid;      // v4 group
            const int row = g & 127;
            const int col = (g >> 7) * 4;
            const int p0  = col >> 1;
            const v4f v = *(const v4f*)(cb + (size_t)(cbase + row) * DIM + col);
            v2f lo; lo.x = v.x; lo.y = v.y;
            v2f hi; hi.x = v.z; hi.y = v.w;
            cbP[(p0 + 0) * CBP_STRIDE + row] = lo;
            cbP[(p0 + 1) * CBP_STRIDE + row] = hi;
        }
        __syncthreads();

        auto loadB = [&](v2f* b, int nt) {
#pragma unroll
            for (int j = 0; j < 16; ++j)
                b[j] = cbP[(2 * j + half) * CBP_STRIDE + nt * 16 + lcol];
        };
        auto tile = [&](const v2f* b, int nt) {
            v8f acc = {0.f, 0.f, 0.f, 0.f, 0.f, 0.f, 0.f, 0.f};
#pragma unroll
            for (int j = 0; j < 16; ++j)
                acc = __builtin_amdgcn_wmma_f32_16x16x4_f32(
                    false, a[j], false, b[j], (short)0, acc, false, false);
            const int   n   = cbase + nt * 16 + lcol;
            const float e2v = e2s[n];
#pragma unroll
            for (int r = 0; r < 8; ++r) {
                const float s = fmaf(-2.f, acc[r], e2v); // ||e||^2 - 2 z.e
                if (s < mv[r]) { mv[r] = s; mi[r] = n; }
            }
        };

        // software-pipelined: B frags for tile nt+1 load during tile nt's WMMAs
        v2f b0[16], b1[16];
        loadB(b0, 0);
#pragma unroll
        for (int nt = 0; nt < NT_PER_CH; nt += 2) {
            loadB(b1, nt + 1);
            tile(b0, nt);
            if (nt + 2 < NT_PER_CH) loadB(b0, nt + 2);
            tile(b1, nt + 1);
        }
    }

    // ---- cross-lane argmin within each 16-lane half (xor 8,4,2,1) ----
#pragma unroll
    for (int off = 8; off >= 1; off >>= 1) {
#pragma unroll
        for (int r = 0; r < 8; ++r) {
            const float ov = __shfl_xor(mv[r], off, 32);
            const int   oi = __shfl_xor(mi[r], off, 32);
            if (ov < mv[r] || (ov == mv[r] && oi < mi[r])) { mv[r] = ov; mi[r] = oi; }
        }
    }

    // ---- gather z_q rows (index is wave-uniform -> readlane/SGPR base) ----
    float lsum = 0.f;
    const size_t obase = (size_t)tokBase * DIM;
#pragma unroll
    for (int m = 0; m < 16; ++m) {
        const int idx = __builtin_amdgcn_readlane(mi[m & 7], (m >> 3) << 4);
        const float* crow = cb + (size_t)idx * DIM + lane * 2;
        const float* zrow = z + obase + (size_t)m * DIM + lane * 2;
        const float c0 = crow[0], c1 = crow[1];
        const float z0 = zrow[0], z1 = zrow[1];
        const float d0 = c0 - z0, d1 = c1 - z1;
        lsum = fmaf(d0, d0, lsum);
        lsum = fmaf(d1, d1, lsum);
        float* orow = out + obase + (size_t)m * DIM + lane * 2;
        orow[0] = c0;
        orow[1] = c1;
    }

    // deterministic reduction: wave -> LDS -> per-WG partial
#pragma unroll
    for (int off = 16; off >= 1; off >>= 1) lsum += __shfl_xor(lsum, off, 32);
    if (lane == 0) wsum[wave] = lsum;
    __syncthreads();
    if (tid == 0) {
        float s = 0.f;
#pragma unroll
        for (int w = 0; w < WAVES_WG; ++w) s += wsum[w];
        partials[blockIdx.x] = s;
    }
}

// --------------------------------------------------------------- finish ---
__global__ __launch_bounds__(1024) void vq_finish(const float* __restrict__ partials,
                                                  float* __restrict__ out_scalar) {
    __shared__ float s[1024];
    const int t = threadIdx.x;
    s[t] = partials[t] + partials[t + 1024];   // 2048 partials, fixed pairing
    __syncthreads();
    for (int off = 512; off >= 1; off >>= 1) {
        if (t < off) s[t] += s[t + off];
        __syncthreads();
    }
    if (t == 0)
        *out_scalar = 1.25f * s[0] / (float)((double)N_TOKENS_C * (double)DIM);
}

// --------------------------------------------------------------- launch ---
extern "C" void kernel_launch(void* const* d_in, const int* in_sizes, int n_in,
                              void* d_out, int out_size, void* d_ws, size_t ws_size,
                              hipStream_t stream) {
    const float* z  = (const float*)d_in[0];   // [262144, 64] fp32
    const float* cb = (const float*)d_in[1];   // [512, 64] fp32
    float* out = (float*)d_out;                // [262144*64] z_q  + [1] loss
    float* ws  = (float*)d_ws;                 // [0..511] e2, [512..2559] partials

    vq_prep  <<<1, 512, 0, stream>>>(cb, ws);
    vq_main  <<<GRID_MAIN, 256, 0, stream>>>(z, cb, ws, out, ws + NUM_E);
    vq_finish<<<1, 1024, 0, stream>>>(ws + NUM_E, out + (size_t)N_TOKENS_C * DIM);
}